// s2v_embedding_65111704208101
// MI455X (gfx1250) — compile-verified
//
#include <hip/hip_runtime.h>

typedef __attribute__((ext_vector_type(2))) float v2f;
typedef __attribute__((ext_vector_type(8))) float v8f;

#define N_NODES 100000
#define N_EDGES 1600000
#define D       64
#define NTILES  (N_NODES / 16)   // 6250 exact

// ---------------------------------------------------------------------------
// K0: zero the neighbor-sum workspace and the 64-float output accumulator.
// ---------------------------------------------------------------------------
__global__ void zero_ws_kernel(float* __restrict__ nbr, float* __restrict__ out) {
    size_t i = (size_t)blockIdx.x * blockDim.x + threadIdx.x;
    const size_t n4 = (size_t)N_NODES * D / 4;   // 1.6M float4
    if (i < n4) {
        float4 z; z.x = 0.f; z.y = 0.f; z.z = 0.f; z.w = 0.f;
        reinterpret_cast<float4*>(nbr)[i] = z;
    }
    if (blockIdx.x == 0 && threadIdx.x < D) out[threadIdx.x] = 0.f;
}

// ---------------------------------------------------------------------------
// K1: nbr_sum[dst] += emb[src]  (segment_sum). 16 threads per edge, each
// handles one float4 => 256B fully-coalesced gather + 4 f32 global atomics.
// ---------------------------------------------------------------------------
__global__ void edge_scatter_kernel(const float* __restrict__ emb,
                                    const int*   __restrict__ esrc,
                                    const int*   __restrict__ edst,
                                    float*       __restrict__ nbr) {
    unsigned gid = blockIdx.x * blockDim.x + threadIdx.x;
    unsigned e   = gid >> 4;
    if (e >= N_EDGES) return;
    unsigned q = (gid & 15u) * 4u;
    int s = esrc[e];
    int d = edst[e];
    const float4 v = *reinterpret_cast<const float4*>(emb + (size_t)s * D + q);
    float* o = nbr + (size_t)d * D + q;
    atomicAdd(o + 0, v.x);
    atomicAdd(o + 1, v.y);
    atomicAdd(o + 2, v.z);
    atomicAdd(o + 3, v.w);
}

// ---------------------------------------------------------------------------
// K2: graph_emb = sum_rows relu(feat@W1^T + b1 + nbr@W2^T + b2)
// One wave (wave32) owns a 16-node tile. fp32 WMMA 16x16x4, K unrolled x16,
// two weight matrices chained into one f32 accumulator per 16-col tile.
// ---------------------------------------------------------------------------
__global__ void __launch_bounds__(256)
fused_gemm_relu_sum_kernel(const float* __restrict__ feat,
                           const float* __restrict__ nbr,
                           const float* __restrict__ W1,
                           const float* __restrict__ b1,
                           const float* __restrict__ W2,
                           const float* __restrict__ b2,
                           float*       __restrict__ out) {
    __shared__ float W1s[D][D + 1];   // +1 pad: kills 16-way LDS bank conflict
    __shared__ float W2s[D][D + 1];
    __shared__ float colsum[D];

    // Cooperative load of W1/W2 (row-major [n][k]) into LDS.
    for (int idx = threadIdx.x; idx < D * D; idx += 256) {
        int r = idx >> 6, c = idx & 63;
        W1s[r][c] = W1[idx];
        W2s[r][c] = W2[idx];
    }
    if (threadIdx.x < D) colsum[threadIdx.x] = 0.f;
    __syncthreads();

    const int lane = threadIdx.x & 31;
    const int wv   = threadIdx.x >> 5;
    const int tile = blockIdx.x * 8 + wv;

    if (tile < NTILES) {   // wave-uniform guard; barriers stay block-uniform
        const int m  = lane & 15;
        const int kh = (lane >> 4) * 2;            // 0 (lanes 0-15) or 2 (16-31)
        const size_t row = (size_t)(tile * 16 + m) * D;

        // A operands: fp32 16x4 WMMA layout, all 16 K-blocks, reused over 4 N-tiles.
        v2f af[16], an[16];
#pragma unroll
        for (int kb = 0; kb < 16; ++kb) {
            const float2 fa = *reinterpret_cast<const float2*>(feat + row + 4 * kb + kh);
            const float2 na = *reinterpret_cast<const float2*>(nbr  + row + 4 * kb + kh);
            af[kb].x = fa.x; af[kb].y = fa.y;
            an[kb].x = na.x; an[kb].y = na.y;
        }

#pragma unroll
        for (int nt = 0; nt < 4; ++nt) {
            const int n = nt * 16 + (lane & 15);   // output column for this lane
            v8f c = {};
            // C += feat_tile @ W1^T   (B element (k,n) = W1[n][k])
#pragma unroll
            for (int kb = 0; kb < 16; ++kb) {
                v2f bv;
                bv.x = W1s[n][4 * kb + kh];
                bv.y = W1s[n][4 * kb + kh + 1];
                c = __builtin_amdgcn_wmma_f32_16x16x4_f32(
                        false, af[kb], false, bv, (short)0, c, false, false);
            }
            // C += nbr_tile @ W2^T
#pragma unroll
            for (int kb = 0; kb < 16; ++kb) {
                v2f bv;
                bv.x = W2s[n][4 * kb + kh];
                bv.y = W2s[n][4 * kb + kh + 1];
                c = __builtin_amdgcn_wmma_f32_16x16x4_f32(
                        false, an[kb], false, bv, (short)0, c, false, false);
            }
            // bias + relu + row reduction (lane holds 8 rows of column n)
            const float bias = b1[n] + b2[n];
            float part = 0.f;
#pragma unroll
            for (int r = 0; r < 8; ++r) part += fmaxf(c[r] + bias, 0.f);
            part += __shfl_xor(part, 16, 32);      // combine rows 0-7 with 8-15
            if (lane < 16) atomicAdd(&colsum[n], part);
        }
    }
    __syncthreads();
    if (threadIdx.x < D) atomicAdd(&out[threadIdx.x], colsum[threadIdx.x]);
}

// ---------------------------------------------------------------------------
extern "C" void kernel_launch(void* const* d_in, const int* in_sizes, int n_in,
                              void* d_out, int out_size, void* d_ws, size_t ws_size,
                              hipStream_t stream) {
    const float* feat = (const float*)d_in[0];
    const float* emb  = (const float*)d_in[1];
    const float* W1   = (const float*)d_in[2];
    const float* b1   = (const float*)d_in[3];
    const float* W2   = (const float*)d_in[4];
    const float* b2   = (const float*)d_in[5];
    const int*   esrc = (const int*)d_in[6];
    const int*   edst = (const int*)d_in[7];
    float* out = (float*)d_out;
    float* nbr = (float*)d_ws;        // N_NODES*D floats = 25.6 MB scratch

    const int zblocks = (N_NODES * D / 4 + 255) / 256;          // 6250
    zero_ws_kernel<<<zblocks, 256, 0, stream>>>(nbr, out);

    const int eblocks = (N_EDGES * 16) / 256;                   // 100000
    edge_scatter_kernel<<<eblocks, 256, 0, stream>>>(emb, esrc, edst, nbr);

    const int gblocks = (NTILES + 7) / 8;                       // 782
    fused_gemm_relu_sum_kernel<<<gblocks, 256, 0, stream>>>(
        feat, nbr, W1, b1, W2, b2, out);
}